// HadamardTransform_9509057593676
// MI455X (gfx1250) — compile-verified
//
#include <hip/hip_runtime.h>

// FWHT over rows of 4096 fp32 = H16 (x) H16 (x) H16, three rounds of 16x16x16
// fp32 WMMA (V_WMMA_F32_16X16X4_F32 chained x4). One workgroup (256 thr = 8
// waves) per row; LDS staging with stride-17 padding for bank-conflict-free
// strided tile access. Memory-bound: 512 MB total -> ~22 us roofline @ 23.3 TB/s.

typedef float v2f __attribute__((ext_vector_type(2)));
typedef float v8f __attribute__((ext_vector_type(8)));

#define PADK 17  // 16 floats per k-row + 1 pad; gcd(17,64)=1 -> conflict-free

__device__ __forceinline__ float hsign(int a, int b) {
    return (__popc(a & b) & 1) ? -1.0f : 1.0f;
}

__global__ __launch_bounds__(256) void fwht4096_wmma_kernel(
    const float* __restrict__ x, float* __restrict__ out)
{
    __shared__ float lds[16 * 16 * PADK];  // 17408 B

    const int row  = blockIdx.x;
    const int tid  = threadIdx.x;      // 0..255
    const int lane = tid & 31;
    const int wave = tid >> 5;         // 0..7
    const int m    = lane & 15;        // matrix row/col index within lane-half
    const int h    = lane >> 4;        // lane half (0/1)

    // H16 sign registers. hreg[q] packs rows K = 4q + {0,1} + 2h, col = m.
    // H16 is symmetric, so the same registers serve as B (stages 1-2, layout
    // B[K][n]: vgpr p, half h -> K = 4q+p+2h, n = m) and as A (stage 3, layout
    // A[M][K]: vgpr p, half h -> M = m, K = 4q+p+2h).
    v2f hreg[4];
#pragma unroll
    for (int q = 0; q < 4; ++q) {
        hreg[q].x = hsign(4 * q + 0 + 2 * h, m);
        hreg[q].y = hsign(4 * q + 1 + 2 * h, m);
    }

    // ---- global -> LDS (padded layout: elem e=(i*16+j)*16+k -> (e>>4)*17 + (e&15)) ----
    const float* src = x + (size_t)row * 4096;
#pragma unroll
    for (int c = 0; c < 4; ++c) {
        int e = c * 1024 + tid * 4;                 // 16B-aligned, within one k-row
        float4 v = *(const float4*)(src + e);
        float* p = &lds[(e >> 4) * PADK + (e & 15)];
        p[0] = v.x; p[1] = v.y; p[2] = v.z; p[3] = v.w;
    }
    __syncthreads();

    // ---- stage 1: transform along k. Tile per i: T[j][k], D = T x H ----
#pragma unroll
    for (int t = 0; t < 2; ++t) {
        const int i    = wave + 8 * t;
        const int base = i * 16 * PADK;
        v8f acc = {0.f, 0.f, 0.f, 0.f, 0.f, 0.f, 0.f, 0.f};
#pragma unroll
        for (int q = 0; q < 4; ++q) {
            v2f a;  // A[m][K=4q+p+2h] = lds[base + m*PADK + K]
            a.x = lds[base + m * PADK + (4 * q + 0 + 2 * h)];
            a.y = lds[base + m * PADK + (4 * q + 1 + 2 * h)];
            acc = __builtin_amdgcn_wmma_f32_16x16x4_f32(
                false, a, false, hreg[q], (short)0, acc, false, false);
        }
#pragma unroll
        for (int v = 0; v < 8; ++v)                 // D[j=v+8h][k'=m]
            lds[base + (v + 8 * h) * PADK + m] = acc[v];
    }
    __syncthreads();

    // ---- stage 2: transform along j. Tile per i: T[k][j] (rowStride 1, colStride PADK) ----
#pragma unroll
    for (int t = 0; t < 2; ++t) {
        const int i    = wave + 8 * t;
        const int base = i * 16 * PADK;
        v8f acc = {0.f, 0.f, 0.f, 0.f, 0.f, 0.f, 0.f, 0.f};
#pragma unroll
        for (int q = 0; q < 4; ++q) {
            v2f a;  // A[m=k][K=j] = lds[base + j*PADK + k]
            a.x = lds[base + (4 * q + 0 + 2 * h) * PADK + m];
            a.y = lds[base + (4 * q + 1 + 2 * h) * PADK + m];
            acc = __builtin_amdgcn_wmma_f32_16x16x4_f32(
                false, a, false, hreg[q], (short)0, acc, false, false);
        }
#pragma unroll
        for (int v = 0; v < 8; ++v)                 // D[k=v+8h][j'=m]
            lds[base + m * PADK + (v + 8 * h)] = acc[v];
    }
    __syncthreads();

    // ---- stage 3: transform along i. D = H x T (H in A) so D cols = contiguous k.
    //      Write directly to global, fused 1/sqrt(4096) scale, 64B-contiguous chunks. ----
    float* dst = out + (size_t)row * 4096;
#pragma unroll
    for (int t = 0; t < 2; ++t) {
        const int j = wave + 8 * t;
        v8f acc = {0.f, 0.f, 0.f, 0.f, 0.f, 0.f, 0.f, 0.f};
#pragma unroll
        for (int q = 0; q < 4; ++q) {
            v2f b;  // B[K=i][n=k] = lds[(i*16+j)*PADK + k]; vgpr p, half h -> i = 4q+p+2h
            b.x = lds[((4 * q + 0 + 2 * h) * 16 + j) * PADK + m];
            b.y = lds[((4 * q + 1 + 2 * h) * 16 + j) * PADK + m];
            acc = __builtin_amdgcn_wmma_f32_16x16x4_f32(
                false, hreg[q], false, b, (short)0, acc, false, false);
        }
#pragma unroll
        for (int v = 0; v < 8; ++v)                 // D[i'=v+8h][k=m]
            dst[(v + 8 * h) * 256 + j * 16 + m] = acc[v] * (1.0f / 64.0f);
    }
}

extern "C" void kernel_launch(void* const* d_in, const int* in_sizes, int n_in,
                              void* d_out, int out_size, void* d_ws, size_t ws_size,
                              hipStream_t stream) {
    const float* x = (const float*)d_in[0];
    float* out = (float*)d_out;
    const int rows = in_sizes[0] / 4096;   // 16384
    fwht4096_wmma_kernel<<<rows, 256, 0, stream>>>(x, out);
}